// UltrametricHopfieldMemory_5016521801933
// MI455X (gfx1250) — compile-verified
//
#include <hip/hip_runtime.h>

#define DIM   1024
#define BF    32
#define DEPTH 3
#define BATCH 2048

typedef __attribute__((ext_vector_type(2))) float v2f;
typedef __attribute__((ext_vector_type(4))) float v4f;
typedef __attribute__((ext_vector_type(8))) float v8f;

// One wave32 per batch element. Scores the 32 children per level with
// V_WMMA_F32_16X16X4_F32. K is relabeled so each half-wave owns 4 contiguous
// floats per 8-K chunk -> b128 global loads for A, b128 LDS loads for B
// (query replicated across all 16 columns). Both 16-row halves run fused in
// one k-loop: shared B loads, two independent WMMA accumulation chains.
__global__ __launch_bounds__(32) void umhm_kernel(
    const float* __restrict__ q,   // (BATCH, DIM)
    const float* __restrict__ t0,  // (1, DIM)
    const float* __restrict__ t1,  // (BF, DIM)
    const float* __restrict__ t2,  // (BF^2, DIM)
    const float* __restrict__ t3,  // (BF^3, DIM)
    float* __restrict__ out_states,// (BATCH, DIM)
    int* __restrict__ out_idx)     // (BATCH,)
{
    __shared__ float qs[DIM];      // query row staged for broadcast B reads

    const int b    = blockIdx.x;
    const int lane = threadIdx.x;          // 0..31

    // Stage query row into LDS, coalesced b128.
    const float* qrow = q + (size_t)b * DIM;
    #pragma unroll
    for (int i = 0; i < DIM / (32 * 4); ++i) {
        const int off = i * 128 + lane * 4;
        *(v4f*)(qs + off) = *(const v4f*)(qrow + off);
    }
    __syncthreads();

    // Running state = root delta; coalesced layout: elem i*128 + lane*4 + j.
    v4f st[8];
    #pragma unroll
    for (int i = 0; i < 8; ++i)
        st[i] = *(const v4f*)(t0 + i * 128 + lane * 4);

    const float* tables[DEPTH] = { t1, t2, t3 };

    const int m   = lane & 15;         // A-matrix row within a 16-row half
    const int kk4 = (lane >> 4) * 4;   // this half-wave's 4 contiguous K elems

    int idx = 0;
    #pragma unroll
    for (int d = 0; d < DEPTH; ++d) {
        const float* __restrict__ T = tables[d];
        const int base = idx * BF;

        const float* __restrict__ row0 = T + (size_t)(base + m)      * DIM;
        const float* __restrict__ row1 = T + (size_t)(base + 16 + m) * DIM;

        v8f acc0 = {};                 // scores for rows 0..15
        v8f acc1 = {};                 // scores for rows 16..31
        #pragma unroll 4
        for (int k0 = 0; k0 < DIM; k0 += 8) {
            const v4f a0 = *(const v4f*)(row0 + k0 + kk4); // A half 0, b128
            const v4f a1 = *(const v4f*)(row1 + k0 + kk4); // A half 1, b128
            const v4f bq = *(const v4f*)(qs   + k0 + kk4); // B chunk, b128 LDS

            const v2f a0lo = __builtin_shufflevector(a0, a0, 0, 1);
            const v2f a0hi = __builtin_shufflevector(a0, a0, 2, 3);
            const v2f a1lo = __builtin_shufflevector(a1, a1, 0, 1);
            const v2f a1hi = __builtin_shufflevector(a1, a1, 2, 3);
            const v2f blo  = __builtin_shufflevector(bq, bq, 0, 1);
            const v2f bhi  = __builtin_shufflevector(bq, bq, 2, 3);

            acc0 = __builtin_amdgcn_wmma_f32_16x16x4_f32(
                       false, a0lo, false, blo, (short)0, acc0, false, false);
            acc1 = __builtin_amdgcn_wmma_f32_16x16x4_f32(
                       false, a1lo, false, blo, (short)0, acc1, false, false);
            acc0 = __builtin_amdgcn_wmma_f32_16x16x4_f32(
                       false, a0hi, false, bhi, (short)0, acc0, false, false);
            acc1 = __builtin_amdgcn_wmma_f32_16x16x4_f32(
                       false, a1hi, false, bhi, (short)0, acc1, false, false);
        }

        // D layout: lanes 0-15 hold M=r in acc[r]; lanes 16-31 hold M=r+8.
        // All 16 columns are identical (B replicated), so each lane sees 16
        // of the 32 scores: rows {h*16 + rowoff + r}.
        const int rowoff = (lane >> 4) * 8;
        float bestv = acc0[0];
        int   besti = rowoff;
        #pragma unroll
        for (int r = 1; r < 8; ++r) {
            const int ri = rowoff + r;
            if (acc0[r] > bestv) { bestv = acc0[r]; besti = ri; }
        }
        #pragma unroll
        for (int r = 0; r < 8; ++r) {
            const int ri = 16 + rowoff + r;
            if (acc1[r] > bestv) { bestv = acc1[r]; besti = ri; }
        }
        // Combine the two half-groups (lane l <-> lane l^16).
        const float ov = __shfl_xor(bestv, 16, 32);
        const int   oi = __shfl_xor(besti, 16, 32);
        if (ov > bestv || (ov == bestv && oi < besti)) { bestv = ov; besti = oi; }

        idx = base + besti;

        // Accumulate chosen delta row into state (coalesced b128 across wave).
        const float* __restrict__ crow = T + (size_t)idx * DIM;
        #pragma unroll
        for (int i = 0; i < 8; ++i)
            st[i] += *(const v4f*)(crow + i * 128 + lane * 4);
    }

    // Write outputs: states (f32), then leaf index bits (i32) in d_out tail.
    float* orow = out_states + (size_t)b * DIM;
    #pragma unroll
    for (int i = 0; i < 8; ++i)
        *(v4f*)(orow + i * 128 + lane * 4) = st[i];
    if (lane == 0) out_idx[b] = idx;
}

extern "C" void kernel_launch(void* const* d_in, const int* in_sizes, int n_in,
                              void* d_out, int out_size, void* d_ws, size_t ws_size,
                              hipStream_t stream) {
    const float* q  = (const float*)d_in[0];
    const float* t0 = (const float*)d_in[1];
    const float* t1 = (const float*)d_in[2];
    const float* t2 = (const float*)d_in[3];
    const float* t3 = (const float*)d_in[4];

    float* out_states = (float*)d_out;
    int*   out_idx    = (int*)((float*)d_out + (size_t)BATCH * DIM);

    (void)in_sizes; (void)n_in; (void)out_size; (void)d_ws; (void)ws_size;

    umhm_kernel<<<BATCH, 32, 0, stream>>>(q, t0, t1, t2, t3, out_states, out_idx);
}